// FocalLoss_48765058679587
// MI455X (gfx1250) — compile-verified
//
#include <hip/hip_runtime.h>

typedef __attribute__((ext_vector_type(2))) float v2f;
typedef __attribute__((ext_vector_type(8))) float v8f;

#define BS   256   // 8 waves of 32
#define NBLK 16    // blocks per image
#define MAXM 64

// Branchless per-anchor focal contribution.
// Clamped index + mask instead of EXEC predication: keeps the whole loop
// body straight-line so the WMMA always executes with EXEC all-ones.
__device__ __forceinline__ float anchor_term(
    int a, int A, int b,
    const float* __restrict__ anchors, const float* __restrict__ cls,
    const float* __restrict__ sS, const float* __restrict__ sE,
    const float* __restrict__ sL, int M, int& posCount)
{
    const bool valid = (a < A);
    const int  ac    = valid ? a : (A - 1);

    float ax = anchors[(size_t)ac * 2];

    // min-width covering annotation; strict '<' reproduces stable-argsort +
    // first-hit semantics (min width, lower original index on ties).
    float bestw = 3.4e38f;
    float flab  = -1.0f;
    for (int j = 0; j < M; ++j) {
        float s = sS[j], e = sE[j], l = sL[j];
        float w = e - s;
        bool take = (ax >= s) & (ax <= e) & (w < bestw);
        bestw = take ? w : bestw;     // v_cndmask, no exec manipulation
        flab  = take ? l : flab;
    }
    const bool positive = valid & (flab >= 0.0f);
    posCount += positive ? 1 : 0;
    const int lab = (int)flab;

    const float4* cp4 = (const float4*)(cls + ((size_t)b * A + ac) * 16);
    float asum = 0.0f;
    #pragma unroll
    for (int q4 = 0; q4 < 4; ++q4) {            // C == 16 -> 4x b128 loads
        float4 v = cp4[q4];
        float pv[4] = {v.x, v.y, v.z, v.w};
        #pragma unroll
        for (int k = 0; k < 4; ++k) {
            int c = q4 * 4 + k;
            float p = fminf(fmaxf(pv[k], 1.0e-4f), 1.0f - 1.0e-4f);
            bool  t = positive & (c == lab);
            // focal term collapses to alpha*(1-q)^2 * (-log q), q = t?p:1-p
            float q     = t ? p : (1.0f - p);
            float alpha = t ? 0.25f : 0.75f;
            float omq   = 1.0f - q;
            asum += alpha * omq * omq * (-__logf(q));
        }
    }
    return valid ? asum : 0.0f;
}

__global__ __launch_bounds__(BS) void focal_main(
    const float* __restrict__ cls,     // [B, A, C]
    const float* __restrict__ anchors, // [1, A, 2]
    const float* __restrict__ ann,     // [B, M, 3]
    float* __restrict__ fpart,         // [B, NBLK]
    int*   __restrict__ ipart,         // [B, NBLK]
    int A, int M)
{
    const int b   = blockIdx.y;
    const int tid = threadIdx.x;

    __shared__ float sS[MAXM], sE[MAXM], sL[MAXM];
    __shared__ float swsum[BS / 32];
    __shared__ int   swpos[BS / 32];

    for (int j = tid; j < M; j += BS) {
        const float* ap = ann + ((size_t)b * M + j) * 3;
        sS[j] = ap[0];
        sE[j] = ap[1];
        sL[j] = ap[2];
    }
    __syncthreads();

    const int stride = NBLK * BS;
    const int pairs  = (A + 2 * stride - 1) / (2 * stride);  // uniform: 7
    const int base   = blockIdx.x * BS + tid;

    v8f acc = {};
    const v2f ones = {1.0f, 1.0f};
    int posCount = 0;

    for (int it = 0; it < pairs; ++it) {
        int a0 = base + (2 * it) * stride;
        int a1 = a0 + stride;

        // prefetch next pair's classification lines (global_prefetch_b8)
        int an = a0 + 2 * stride;
        an = an < A ? an : (A - 1);
        __builtin_prefetch(cls + ((size_t)b * A + an) * 16, 0, 0);

        v2f av;
        av.x = anchor_term(a0, A, b, anchors, cls, sS, sE, sL, M, posCount);
        av.y = anchor_term(a1, A, b, anchors, cls, sS, sE, sL, M, posCount);

        // D = A x ones + C : accumulates all 64 per-lane partials per issue
        acc = __builtin_amdgcn_wmma_f32_16x16x4_f32(
                  false, av, false, ones, (short)0, acc, false, false);
    }

    // Every D column holds a row-sum; lanes L and L^16 together cover M=0..15,
    // so s + shfl_xor(s,16) is the full wave total in every lane.
    float s = acc[0] + acc[1] + acc[2] + acc[3] + acc[4] + acc[5] + acc[6] + acc[7];
    s += __shfl_xor(s, 16, 32);
    for (int off = 16; off > 0; off >>= 1)
        posCount += __shfl_xor(posCount, off, 32);

    int wave = tid >> 5, lane = tid & 31;
    if (lane == 0) { swsum[wave] = s; swpos[wave] = posCount; }
    __syncthreads();
    if (tid == 0) {
        float bsum = 0.0f; int bpos = 0;
        for (int w = 0; w < BS / 32; ++w) { bsum += swsum[w]; bpos += swpos[w]; }
        fpart[b * NBLK + blockIdx.x] = bsum;
        ipart[b * NBLK + blockIdx.x] = bpos;
    }
}

// Deterministic finalize: fixed-order sums, per-image /max(num_pos,1).
__global__ void focal_finalize(const float* __restrict__ fpart,
                               const int*   __restrict__ ipart,
                               float* __restrict__ out, int B)
{
    if (threadIdx.x == 0 && blockIdx.x == 0) {
        float total = 0.0f;
        for (int b = 0; b < B; ++b) {
            float s = 0.0f; int np = 0;
            for (int k = 0; k < NBLK; ++k) {
                s  += fpart[b * NBLK + k];
                np += ipart[b * NBLK + k];
            }
            total += s / (float)(np > 0 ? np : 1);
        }
        out[0] = total;
    }
}

extern "C" void kernel_launch(void* const* d_in, const int* in_sizes, int n_in,
                              void* d_out, int out_size, void* d_ws, size_t ws_size,
                              hipStream_t stream) {
    const float* cls = (const float*)d_in[0];
    const float* anc = (const float*)d_in[1];
    const float* ann = (const float*)d_in[2];

    const int C = 16;
    int A = in_sizes[1] / 2;                 // anchors: (1, A, 2)
    int B = in_sizes[0] / (A * C);           // classifications: (B, A, C)
    int M = in_sizes[2] / (B * 3);           // annotations: (B, M, 3)

    float* fpart = (float*)d_ws;
    int*   ipart = (int*)((char*)d_ws + (size_t)B * NBLK * sizeof(float));

    dim3 grid(NBLK, B);
    focal_main<<<grid, BS, 0, stream>>>(cls, anc, ann, fpart, ipart, A, M);
    focal_finalize<<<1, 32, 0, stream>>>(fpart, ipart, (float*)d_out, B);
}